// Center_16217796510058
// MI455X (gfx1250) — compile-verified
//
#include <hip/hip_runtime.h>
#include <stdint.h>

// Center-update op: out = centers; out[labels[b]] += (ALPHA-1)*(centers[labels[b]] - features[b])
// Pure data movement + scatter-add: ~400 MB traffic -> ~17us at 23.3 TB/s. No WMMA applies.
// CDNA5 features used: async global<->LDS DMA (ASYNCcnt) for the bulk copy, and
// hardware f32 global atomics for the duplicate-accumulating scatter.
//
// NOTE: center_scatter is defined FIRST so its atomics show up in the asm snippet
// (verifying global_atomic_add_f32 vs. a CAS loop).

#define DIM 256
static constexpr float SCALE = (float)(0.9 - 1.0);   // (alpha - 1), matches reference f32 constant

// ---------------------------------------------------------------------------
// Phase 2 kernel: gather + scale + scatter-add with hardware f32 atomics.
// 64 threads per batch row; each thread owns one float4 column slice.
// ---------------------------------------------------------------------------
__global__ __launch_bounds__(256) void center_scatter(const float* __restrict__ features,
                                                      const int* __restrict__ labels,
                                                      const float* __restrict__ centers,
                                                      float* __restrict__ out,
                                                      int B) {
    const int t   = blockIdx.x * 256 + threadIdx.x;
    const int row = t >> 6;                      // batch index
    if (row >= B) return;
    const int col = (t & 63) << 2;               // 0,4,...,252

    const int label = labels[row];
    const size_t foff = (size_t)row   * DIM + col;
    const size_t coff = (size_t)label * DIM + col;

    const float4 f = *(const float4*)(features + foff);
    const float4 c = *(const float4*)(centers  + coff);

    // diff = (alpha-1) * (centers[label] - features[row])  -- same assoc. as reference
    const float dx = SCALE * (c.x - f.x);
    const float dy = SCALE * (c.y - f.y);
    const float dz = SCALE * (c.z - f.z);
    const float dw = SCALE * (c.w - f.w);

    float* o = out + coff;
    __hip_atomic_fetch_add(o + 0, dx, __ATOMIC_RELAXED, __HIP_MEMORY_SCOPE_AGENT);
    __hip_atomic_fetch_add(o + 1, dy, __ATOMIC_RELAXED, __HIP_MEMORY_SCOPE_AGENT);
    __hip_atomic_fetch_add(o + 2, dz, __ATOMIC_RELAXED, __HIP_MEMORY_SCOPE_AGENT);
    __hip_atomic_fetch_add(o + 3, dw, __ATOMIC_RELAXED, __HIP_MEMORY_SCOPE_AGENT);
}

// ---------------------------------------------------------------------------
// Phase 1 kernel: copy centers -> out through the CDNA5 async-LDS DMA path.
// Each block moves 512 float4 (8 KB): thread t handles chunk t and chunk t+256.
// ---------------------------------------------------------------------------
__global__ __launch_bounds__(256) void copy_centers_async(const float* __restrict__ src,
                                                          float* __restrict__ dst,
                                                          long total4 /* float4 count */) {
    __shared__ float4 lbuf[512];                    // 8 KB staging buffer
    const uint32_t t = threadIdx.x;

    // LDS byte offsets for this thread's two staging slots (low 32 bits of a
    // generic pointer into LDS are the LDS offset; also forces the allocation).
    const uint32_t l0 = (uint32_t)(uintptr_t)(&lbuf[0]) + t * 16u;
    const uint32_t l1 = l0 + 256u * 16u;

    const long idx0 = (long)blockIdx.x * 512 + (long)t;        // float4 index
    const long idx1 = idx0 + 256;
    const bool a = idx0 < total4;
    const bool b = idx1 < total4;

    const uint64_t g0 = (uint64_t)(uintptr_t)src + (uint64_t)idx0 * 16u;
    const uint64_t g1 = (uint64_t)(uintptr_t)src + (uint64_t)idx1 * 16u;
    const uint64_t d0 = (uint64_t)(uintptr_t)dst + (uint64_t)idx0 * 16u;
    const uint64_t d1 = (uint64_t)(uintptr_t)dst + (uint64_t)idx1 * 16u;

    // Async DMA: global -> LDS (tracked by ASYNCcnt, no VGPR data path)
    if (a) {
        asm volatile("global_load_async_to_lds_b128 %0, %1, off"
                     :: "v"(l0), "v"(g0) : "memory");
    }
    if (b) {
        asm volatile("global_load_async_to_lds_b128 %0, %1, off"
                     :: "v"(l1), "v"(g1) : "memory");
    }
    asm volatile("s_wait_asynccnt 0" ::: "memory");
    // Async DMA: LDS -> global
    if (a) {
        asm volatile("global_store_async_from_lds_b128 %0, %1, off"
                     :: "v"(d0), "v"(l0) : "memory");
    }
    if (b) {
        asm volatile("global_store_async_from_lds_b128 %0, %1, off"
                     :: "v"(d1), "v"(l1) : "memory");
    }
    // stores drain before S_ENDPGM's implicit wait-idle
}

extern "C" void kernel_launch(void* const* d_in, const int* in_sizes, int n_in,
                              void* d_out, int out_size, void* d_ws, size_t ws_size,
                              hipStream_t stream) {
    const float* features = (const float*)d_in[0];   // [B, 256] f32
    const int*   labels   = (const int*)d_in[1];     // [B] int
    const float* centers  = (const float*)d_in[2];   // [N, 256] f32
    float*       out      = (float*)d_out;           // [N, 256] f32

    const int  B      = in_sizes[1];
    const long total4 = (long)out_size / 4;          // out_size = N*256, divisible by 4

    // Phase 1: out = centers (async-LDS DMA copy), 512 float4 per block
    const int copy_blocks = (int)((total4 + 511) / 512);
    copy_centers_async<<<copy_blocks, 256, 0, stream>>>(centers, out, total4);

    // Phase 2: scatter-add (4 rows per 256-thread block)
    const int scat_blocks = (B * 64 + 255) / 256;
    center_scatter<<<scat_blocks, 256, 0, stream>>>(features, labels, centers, out, B);
}